// ISAB_VQVAE_18700287606890
// MI455X (gfx1250) — compile-verified
//
#include <hip/hip_runtime.h>
#include <hip/hip_bf16.h>

#define Bn    128
#define Nn    5000
#define DINn  3
#define DHn   128
#define HEADSn 4
#define INDSn 32
#define EDn   3
#define Kn    1024

// sqrt(1/128)
#define INVS 0.08838834764831845f

typedef float v2f __attribute__((ext_vector_type(2)));
typedef float v8f __attribute__((ext_vector_type(8)));

static __device__ inline v8f vzero8() {
  v8f v;
#pragma unroll
  for (int i = 0; i < 8; ++i) v[i] = 0.0f;
  return v;
}

// D = A(16x4 f32) * B(4x16 f32) + C  -- exact f32 WMMA
static __device__ inline v8f wmma4(v2f a, v2f b, v8f c) {
  return __builtin_amdgcn_wmma_f32_16x16x4_f32(false, a, false, b, (short)0, c, false, false);
}

// ---------------------------------------------------------------------------
// Kernel 0: batch-invariant prep for MAB0.
//   Q[32][128] = I @ Wq0.T + bq0
//   QC[h][q][0..2] = INVS * (Q_h[q] @ Wk0_h)  (score coeffs), [3] = INVS*(Q_h[q]·bk0_h)
// ---------------------------------------------------------------------------
__global__ void k0_prep(const float* __restrict__ I, const float* __restrict__ Wq0,
                        const float* __restrict__ bq0, const float* __restrict__ Wk0,
                        const float* __restrict__ bk0, float* __restrict__ QP,
                        float* __restrict__ QC) {
  __shared__ float Qs[INDSn * 132];
  int t = threadIdx.x;  // 256
  for (int e = t; e < INDSn * DHn; e += 256) {
    int q = e >> 7, i = e & 127;
    float s = bq0[i];
    const float* wr = Wq0 + i * DHn;
    const float* ir = I + q * DHn;
    for (int d = 0; d < DHn; ++d) s += ir[d] * wr[d];
    QP[e] = s;
    Qs[q * 132 + i] = s;
  }
  __syncthreads();
  for (int e = t; e < HEADSn * INDSn * 4; e += 256) {
    int j = e & 3, q = (e >> 2) & 31, h = e >> 7;
    float s = 0.0f;
    for (int d = 0; d < 32; ++d) {
      int i = h * 32 + d;
      float w = (j < 3) ? Wk0[i * DINn + j] : bk0[i];
      s += Qs[q * 132 + i] * w;
    }
    QC[e] = s * INVS;
  }
}

// ---------------------------------------------------------------------------
// Kernel 1: MAB0 -> H[b][32][128]. One block per batch, 256 threads.
// Streaming online-softmax over N with 3-vector accumulators; N split 2-ways.
// ---------------------------------------------------------------------------
__global__ void __launch_bounds__(256) k1_mab0(
    const float* __restrict__ x, const float* __restrict__ QP,
    const float* __restrict__ QC, const float* __restrict__ Wv0,
    const float* __restrict__ bv0, const float* __restrict__ Wo0,
    const float* __restrict__ bo0, float* __restrict__ Hout) {
  int b = blockIdx.x;
  int t = threadIdx.x;
  __shared__ float xs[3][1024];
  __shared__ float sm_m[256], sm_l[256], sm_a0[256], sm_a1[256], sm_a2[256];
  __shared__ float Olds[INDSn * 132];

  int hq = t & 127, half = t >> 7;
  int h = hq >> 5, q = hq & 31;
  float c0 = QC[hq * 4 + 0], c1 = QC[hq * 4 + 1], c2 = QC[hq * 4 + 2], cc = QC[hq * 4 + 3];
  float m = -1e30f, l = 0.0f, a0 = 0.0f, a1 = 0.0f, a2 = 0.0f;
  const float* xb = x + (size_t)b * DINn * Nn;

  for (int base = 0; base < Nn; base += 1024) {
    int cn = Nn - base;
    if (cn > 1024) cn = 1024;
    __syncthreads();
    for (int e = t; e < 3 * 1024; e += 256) {
      int c = e >> 10, i = e & 1023;
      xs[c][i] = (i < cn) ? xb[c * Nn + base + i] : 0.0f;
    }
    __syncthreads();
    for (int i = half; i < cn; i += 2) {
      float X0 = xs[0][i], X1 = xs[1][i], X2 = xs[2][i];
      float s = c0 * X0 + c1 * X1 + c2 * X2 + cc;
      if (s > m) {
        float r = __expf(m - s);
        l *= r; a0 *= r; a1 *= r; a2 *= r; m = s;
      }
      float w = __expf(s - m);
      l += w; a0 += w * X0; a1 += w * X1; a2 += w * X2;
    }
  }
  sm_m[t] = m; sm_l[t] = l; sm_a0[t] = a0; sm_a1[t] = a1; sm_a2[t] = a2;
  __syncthreads();
  if (t < 128) {
    float m2 = sm_m[t + 128];
    float M = fmaxf(m, m2);
    float r1 = __expf(m - M), r2 = __expf(m2 - M);
    l = l * r1 + sm_l[t + 128] * r2;
    a0 = a0 * r1 + sm_a0[t + 128] * r2;
    a1 = a1 * r1 + sm_a1[t + 128] * r2;
    a2 = a2 * r1 + sm_a2[t + 128] * r2;
    float inv = 1.0f / l;
    a0 *= inv; a1 *= inv; a2 *= inv;
    for (int d = 0; d < 32; ++d) {
      int i = h * 32 + d;
      float o = QP[q * 128 + i] + Wv0[i * 3 + 0] * a0 + Wv0[i * 3 + 1] * a1 +
                Wv0[i * 3 + 2] * a2 + bv0[i];
      Olds[q * 132 + i] = o;
    }
  }
  __syncthreads();
  // H = O + relu(O @ Wo0.T + bo0)
  float* Hb = Hout + (size_t)b * INDSn * DHn;
  for (int e = t; e < INDSn * DHn; e += 256) {
    int q2 = e >> 7, i = e & 127;
    float s = bo0[i];
    const float* wr = Wo0 + i * DHn;
    for (int j = 0; j < DHn; ++j) s += Olds[q2 * 132 + j] * wr[j];
    Hb[e] = Olds[q2 * 132 + i] + fmaxf(s, 0.0f);
  }
}

// ---------------------------------------------------------------------------
// Kernel 2: MAB1 prep per batch: Kx=H@Wk1.T+bk1, V=H@Wv1.T+bv1,
//   KC[h][k][0..2] = INVS*(K_h[k] @ Wq1_h), [3]=INVS*(K_h[k]·bq1_h)
//   VH[h][k][d] = V[k][h*32+d]
// ---------------------------------------------------------------------------
__global__ void __launch_bounds__(256) k2_prep1(
    const float* __restrict__ Hin, const float* __restrict__ Wk1,
    const float* __restrict__ bk1, const float* __restrict__ Wv1,
    const float* __restrict__ bv1, const float* __restrict__ Wq1,
    const float* __restrict__ bq1, float* __restrict__ KC, float* __restrict__ VH) {
  int b = blockIdx.x, t = threadIdx.x;
  __shared__ float Hs[INDSn * 132];
  __shared__ float Kxs[INDSn * 132];
  const float* Hb = Hin + (size_t)b * INDSn * DHn;
  for (int e = t; e < INDSn * DHn; e += 256) Hs[(e >> 7) * 132 + (e & 127)] = Hb[e];
  __syncthreads();
  float* VHb = VH + (size_t)b * HEADSn * 32 * 32;
  for (int e = t; e < INDSn * DHn; e += 256) {
    int k = e >> 7, i = e & 127;
    float skx = bk1[i], sv = bv1[i];
    const float* wk = Wk1 + i * DHn;
    const float* wv = Wv1 + i * DHn;
    for (int d = 0; d < DHn; ++d) {
      float hv = Hs[k * 132 + d];
      skx += hv * wk[d];
      sv += hv * wv[d];
    }
    Kxs[k * 132 + i] = skx;
    int h = i >> 5, dd = i & 31;
    VHb[(h * 32 + k) * 32 + dd] = sv;
  }
  __syncthreads();
  float* KCb = KC + (size_t)b * HEADSn * 32 * 4;
  for (int e = t; e < HEADSn * 32 * 4; e += 256) {
    int j = e & 3, k = (e >> 2) & 31, h = e >> 7;
    float s = 0.0f;
    for (int d = 0; d < 32; ++d) {
      int i = h * 32 + d;
      float w = (j < 3) ? Wq1[i * 3 + j] : bq1[i];
      s += Kxs[k * 132 + i] * w;
    }
    KCb[e] = s * INVS;
  }
}

// ---------------------------------------------------------------------------
// Kernel 3: MAB1 main. Block = 64 points of one batch, 256 threads (8 waves).
// Phase B: per-point attention into O tile (LDS). Phase C: WMMA f32 GEMM
// R = O @ Wo1.T, Y = O + relu(R + bo1), accumulate column sums -> YS[b][128].
// ---------------------------------------------------------------------------
#define MT3 64
__global__ void __launch_bounds__(256) k3_mab1(
    const float* __restrict__ x, const float* __restrict__ KC,
    const float* __restrict__ VH, const float* __restrict__ Wq1,
    const float* __restrict__ bq1, const float* __restrict__ Wo1,
    const float* __restrict__ bo1, float* __restrict__ YS) {
  int b = blockIdx.y;
  int n0 = blockIdx.x * MT3;
  int valid = Nn - n0; if (valid > MT3) valid = MT3;
  int t = threadIdx.x;

  __shared__ float Os[MT3 * 132];     // O tile, padded
  __shared__ float Vhs[HEADSn * 32 * 32];
  __shared__ float Kcs[HEADSn * 32 * 4];
  __shared__ float wqs[128 * 4];      // Wq1 | bq1 packed
  __shared__ float bo1s[128];
  __shared__ float Ysl[128];

  const float* VHb = VH + (size_t)b * 4096;
  for (int e = t; e < 4096; e += 256) Vhs[e] = VHb[e];
  const float* KCb = KC + (size_t)b * 512;
  for (int e = t; e < 512; e += 256) Kcs[e] = KCb[e];
  for (int e = t; e < 512; e += 256) {
    int i = e >> 2, j = e & 3;
    wqs[e] = (j < 3) ? Wq1[i * 3 + j] : bq1[i];
  }
  if (t < 128) { bo1s[t] = bo1[t]; Ysl[t] = 0.0f; }
  __syncthreads();

  // ---- attention: 4 threads per point (one head each) ----
  {
    int p = t >> 2, h = t & 3;
    const float* xb = x + (size_t)b * DINn * Nn + n0;
    if (p < valid) {
      float X0 = xb[p], X1 = xb[Nn + p], X2 = xb[2 * Nn + p];
      float w[32];
      float m = -1e30f;
      const float* kc = &Kcs[h * 128];
#pragma unroll
      for (int k = 0; k < 32; ++k) {
        float s = kc[k * 4] * X0 + kc[k * 4 + 1] * X1 + kc[k * 4 + 2] * X2 + kc[k * 4 + 3];
        w[k] = s;
        m = fmaxf(m, s);
      }
      float l = 0.0f;
#pragma unroll
      for (int k = 0; k < 32; ++k) { float e2 = __expf(w[k] - m); w[k] = e2; l += e2; }
      float inv = 1.0f / l;
      for (int d = 0; d < 32; ++d) {
        float acc = 0.0f;
        const float* vv = &Vhs[h * 1024 + d];
#pragma unroll
        for (int k = 0; k < 32; ++k) acc += w[k] * vv[k * 32];
        int i = h * 32 + d;
        float qv = wqs[i * 4] * X0 + wqs[i * 4 + 1] * X1 + wqs[i * 4 + 2] * X2 + wqs[i * 4 + 3];
        Os[p * 132 + i] = qv + acc * inv;
      }
    } else {
      for (int d = 0; d < 32; ++d) Os[p * 132 + h * 32 + d] = 0.0f;
    }
  }
  __syncthreads();

  // ---- WMMA GEMM: wave w owns N-tile w (cols w*16..+15), 4 M-tiles ----
  int w = t >> 5, lane = t & 31, lm = lane & 15, half = lane >> 4;
  int ncol = w * 16 + lm;
  const float* worow = Wo1 + (size_t)ncol * 128;  // B[k][ncol] = Wo1[ncol][k]
  v8f accs[4];
#pragma unroll
  for (int i = 0; i < 4; ++i) accs[i] = vzero8();
  for (int kb = 0; kb < 128; kb += 4) {
    int k0i = kb + 2 * half;
    v2f bf;
    bf.x = worow[k0i];
    bf.y = worow[k0i + 1];
#pragma unroll
    for (int mt = 0; mt < 4; ++mt) {
      v2f a;
      a.x = Os[(mt * 16 + lm) * 132 + k0i];
      a.y = Os[(mt * 16 + lm) * 132 + k0i + 1];
      accs[mt] = wmma4(a, bf, accs[mt]);
    }
  }
  float ys = 0.0f;
  float bcol = bo1s[ncol];
#pragma unroll
  for (int mt = 0; mt < 4; ++mt) {
#pragma unroll
    for (int r = 0; r < 8; ++r) {
      int row = mt * 16 + r + 8 * half;
      if (row < valid) {
        float y = Os[row * 132 + ncol] + fmaxf(accs[mt][r] + bcol, 0.0f);
        ys += y;
      }
    }
  }
  atomicAdd(&Ysl[ncol], ys);
  __syncthreads();
  if (t < 128) atomicAdd(&YS[b * 128 + t], Ysl[t]);
}

// ---------------------------------------------------------------------------
// Kernel 4: z = (YS/N) @ We.T + be ; VQ argmin ; qloss ; write z + qloss to out.
// ---------------------------------------------------------------------------
__global__ void k4_vq(const float* __restrict__ YS, const float* __restrict__ We,
                      const float* __restrict__ be, const float* __restrict__ cb,
                      float* __restrict__ zq, float* __restrict__ out) {
  __shared__ float Wes[3 * 128];
  __shared__ float red[128];
  int b = threadIdx.x;  // 128
  for (int e = b; e < 384; e += 128) Wes[e] = We[e];
  __syncthreads();
  float z0 = be[0], z1 = be[1], z2 = be[2];
  const float invN = 1.0f / (float)Nn;
  for (int i = 0; i < 128; ++i) {
    float v = YS[b * 128 + i] * invN;
    z0 += v * Wes[i];
    z1 += v * Wes[128 + i];
    z2 += v * Wes[256 + i];
  }
  int best = 0;
  float bd = 1e30f;
  for (int k = 0; k < Kn; ++k) {
    float d0 = z0 - cb[k * 3], d1 = z1 - cb[k * 3 + 1], d2 = z2 - cb[k * 3 + 2];
    float d = d0 * d0 + d1 * d1 + d2 * d2;
    if (d < bd) { bd = d; best = k; }
  }
  float q0 = cb[best * 3], q1 = cb[best * 3 + 1], q2 = cb[best * 3 + 2];
  zq[b * 3 + 0] = q0; zq[b * 3 + 1] = q1; zq[b * 3 + 2] = q2;
  size_t zoff = (size_t)Bn * Nn * DINn + 1;
  out[zoff + b * 3 + 0] = z0;
  out[zoff + b * 3 + 1] = z1;
  out[zoff + b * 3 + 2] = z2;
  float d0 = q0 - z0, d1 = q1 - z1, d2 = q2 - z2;
  red[b] = d0 * d0 + d1 * d1 + d2 * d2;
  __syncthreads();
  for (int s = 64; s > 0; s >>= 1) {
    if (b < s) red[b] += red[b + s];
    __syncthreads();
  }
  if (b == 0) out[(size_t)Bn * Nn * DINn] = 1.25f * red[0] / (float)(Bn * EDn);
}

// ---------------------------------------------------------------------------
// Kernel 5a: decoder front: h2 = lrelu(lrelu(zq@Wd1.T+bd1)@Wd2.T+bd2). Block/batch.
// ---------------------------------------------------------------------------
__global__ void k5a_dec(const float* __restrict__ zq, const float* __restrict__ Wd1,
                        const float* __restrict__ bd1, const float* __restrict__ Wd2,
                        const float* __restrict__ bd2, float* __restrict__ h2) {
  int b = blockIdx.x, t = threadIdx.x;  // 128 threads
  __shared__ float h1s[128];
  float z0 = zq[b * 3], z1 = zq[b * 3 + 1], z2 = zq[b * 3 + 2];
  float v = bd1[t] + Wd1[t * 3] * z0 + Wd1[t * 3 + 1] * z1 + Wd1[t * 3 + 2] * z2;
  h1s[t] = (v > 0.0f) ? v : 0.001f * v;
  __syncthreads();
  if (t < 64) {
    float s = bd2[t];
    const float* wr = Wd2 + t * 128;
    for (int i = 0; i < 128; ++i) s += h1s[i] * wr[i];
    h2[b * 64 + t] = (s > 0.0f) ? s : 0.001f * s;
  }
}

// ---------------------------------------------------------------------------
// Kernel 5b: x_recon = h2 @ Wd3.T + bd3 via WMMA f32. Block = 128 output cols.
// ---------------------------------------------------------------------------
__global__ void __launch_bounds__(256) k5b_recon(const float* __restrict__ h2,
                                                 const float* __restrict__ Wd3,
                                                 const float* __restrict__ bd3,
                                                 float* __restrict__ out) {
  int t = threadIdx.x;
  __shared__ float h2s[128 * 68];
  for (int e = t; e < 128 * 64; e += 256) h2s[(e >> 6) * 68 + (e & 63)] = h2[e];
  __syncthreads();
  int w = t >> 5, lane = t & 31, lm = lane & 15, half = lane >> 4;
  int n = blockIdx.x * 128 + w * 16 + lm;
  bool okn = (n < Nn * DINn);
  float bn = okn ? bd3[n] : 0.0f;
  v2f bfa[16];
#pragma unroll
  for (int s = 0; s < 16; ++s) {
    int k0i = 4 * s + 2 * half;
    bfa[s].x = okn ? Wd3[(size_t)n * 64 + k0i] : 0.0f;
    bfa[s].y = okn ? Wd3[(size_t)n * 64 + k0i + 1] : 0.0f;
  }
#pragma unroll
  for (int mt = 0; mt < 8; ++mt) {
    v8f acc = vzero8();
#pragma unroll
    for (int s = 0; s < 16; ++s) {
      int k0i = 4 * s + 2 * half;
      v2f a;
      a.x = h2s[(mt * 16 + lm) * 68 + k0i];
      a.y = h2s[(mt * 16 + lm) * 68 + k0i + 1];
      acc = wmma4(a, bfa[s], acc);
    }
    if (okn) {
#pragma unroll
      for (int r = 0; r < 8; ++r) {
        int row = mt * 16 + r + 8 * half;
        out[(size_t)row * (Nn * DINn) + n] = acc[r] + bn;
      }
    }
  }
}

// ---------------------------------------------------------------------------
extern "C" void kernel_launch(void* const* d_in, const int* in_sizes, int n_in,
                              void* d_out, int out_size, void* d_ws, size_t ws_size,
                              hipStream_t stream) {
  (void)in_sizes; (void)n_in; (void)out_size; (void)ws_size;
  const float* x    = (const float*)d_in[0];
  const float* I    = (const float*)d_in[1];
  const float* Wq0  = (const float*)d_in[2];
  const float* bq0  = (const float*)d_in[3];
  const float* Wk0  = (const float*)d_in[4];
  const float* bk0  = (const float*)d_in[5];
  const float* Wv0  = (const float*)d_in[6];
  const float* bv0  = (const float*)d_in[7];
  const float* Wo0  = (const float*)d_in[8];
  const float* bo0  = (const float*)d_in[9];
  const float* Wq1  = (const float*)d_in[10];
  const float* bq1  = (const float*)d_in[11];
  const float* Wk1  = (const float*)d_in[12];
  const float* bk1  = (const float*)d_in[13];
  const float* Wv1  = (const float*)d_in[14];
  const float* bv1  = (const float*)d_in[15];
  const float* Wo1  = (const float*)d_in[16];
  const float* bo1  = (const float*)d_in[17];
  const float* We   = (const float*)d_in[18];
  const float* be   = (const float*)d_in[19];
  const float* cbk  = (const float*)d_in[20];
  const float* Wd1  = (const float*)d_in[21];
  const float* bd1  = (const float*)d_in[22];
  const float* Wd2  = (const float*)d_in[23];
  const float* bd2  = (const float*)d_in[24];
  const float* Wd3  = (const float*)d_in[25];
  const float* bd3  = (const float*)d_in[26];

  float* out = (float*)d_out;
  float* W   = (float*)d_ws;
  float* QP = W;                 // 4096
  float* QC = W + 4096;          // 512
  float* H  = W + 4608;          // 524288
  float* KC = W + 528896;        // 65536
  float* VH = W + 594432;        // 524288
  float* YS = W + 1118720;       // 16384
  float* ZQ = W + 1135104;       // 384
  float* H2 = W + 1135488;       // 8192

  hipMemsetAsync(YS, 0, 16384 * sizeof(float), stream);
  k0_prep<<<1, 256, 0, stream>>>(I, Wq0, bq0, Wk0, bk0, QP, QC);
  k1_mab0<<<Bn, 256, 0, stream>>>(x, QP, QC, Wv0, bv0, Wo0, bo0, H);
  k2_prep1<<<Bn, 256, 0, stream>>>(H, Wk1, bk1, Wv1, bv1, Wq1, bq1, KC, VH);
  dim3 g3((Nn + MT3 - 1) / MT3, Bn);
  k3_mab1<<<g3, 256, 0, stream>>>(x, KC, VH, Wq1, bq1, Wo1, bo1, YS);
  k4_vq<<<1, 128, 0, stream>>>(YS, We, be, cbk, ZQ, out);
  k5a_dec<<<Bn, 128, 0, stream>>>(ZQ, Wd1, bd1, Wd2, bd2, H2);
  k5b_recon<<<(Nn * DINn + 127) / 128, 256, 0, stream>>>(H2, Wd3, bd3, out);
}